// GCNPredictor_18975165514587
// MI455X (gfx1250) — compile-verified
//
#include <hip/hip_runtime.h>
#include <hip/hip_bf16.h>

#define NN 512           // num nodes / features
#define OUTSZ (NN * NN)  // 262144

typedef __attribute__((ext_vector_type(2))) float v2f;
typedef __attribute__((ext_vector_type(4))) float v4f;
typedef __attribute__((ext_vector_type(8))) float v8f;

// ---------------- small utility kernels ----------------

__global__ void gcn_fill_f32(float* __restrict__ p, int n, float val) {
    int i = blockIdx.x * blockDim.x + threadIdx.x;
    if (i < n) p[i] = val;
}

// deg[dst] += 1 per edge (deg buffer pre-filled with 1.0 == self loop)
__global__ void gcn_deg_accum(const int* __restrict__ dst, float* __restrict__ deg, int E) {
    int e = blockIdx.x * blockDim.x + threadIdx.x;
    if (e < E) atomicAdd(&deg[dst[e]], 1.0f);
}

__global__ void gcn_rsqrt(float* __restrict__ d, int n) {
    int i = blockIdx.x * blockDim.x + threadIdx.x;
    if (i < n) d[i] = rsqrtf(d[i]);   // deg >= 1 always (self loop)
}

// Dense normalized adjacency: A[d][s] += dis[s]*dis[d]
__global__ void gcn_scatterA(const int* __restrict__ src, const int* __restrict__ dst,
                             const float* __restrict__ dis, float* __restrict__ Amat, int E) {
    int e = blockIdx.x * blockDim.x + threadIdx.x;
    if (e < E) {
        int s = src[e], d = dst[e];
        atomicAdd(&Amat[d * NN + s], dis[s] * dis[d]);
    }
}

__global__ void gcn_diagA(const float* __restrict__ dis, float* __restrict__ Amat) {
    int i = blockIdx.x * blockDim.x + threadIdx.x;
    if (i < NN) Amat[i * NN + i] += dis[i] * dis[i];
}

// ---------------- f32 WMMA GEMM: C[M,N] = A[M,K] @ B[K,N] (+bias, relu) ----------------
// One wave (32 threads) per 16x16 tile of C. V_WMMA_F32_16X16X4_F32 per 4-wide K step.
template <int M, int Nc, int K, bool RELU, bool HAS_BIAS>
__global__ void wmma_gemm_f32(const float* __restrict__ A, const float* __restrict__ B,
                              const float* __restrict__ bias, float* __restrict__ C) {
    constexpr int tilesN = Nc / 16;
    const int tile = blockIdx.x;
    const int tm = (tile / tilesN) * 16;
    const int tn = (tile % tilesN) * 16;
    const int lane = threadIdx.x;     // 0..31, wave32
    const int half = lane >> 4;       // 0 or 1
    const int idx  = lane & 15;       // row (A,C) or col (B,C) within tile

    v8f c;
    const float binit = HAS_BIAS ? bias[tn + idx] : 0.0f;
#pragma unroll
    for (int r = 0; r < 8; ++r) c[r] = binit;

    const float* arow = A + (size_t)(tm + idx) * K;
    for (int k = 0; k < K; k += 4) {
        // A frag (16x4): lane holds A[m][k+2h], A[m][k+2h+1]
        v2f a;
        a[0] = arow[k + 2 * half];
        a[1] = arow[k + 2 * half + 1];
        // B frag (4x16): lane holds B[k+2h][n], B[k+2h+1][n]
        v2f b;
        b[0] = B[(size_t)(k + 2 * half) * Nc + tn + idx];
        b[1] = B[(size_t)(k + 2 * half + 1) * Nc + tn + idx];
        c = __builtin_amdgcn_wmma_f32_16x16x4_f32(false, a, false, b, (short)0, c, false, false);
    }

    // C/D layout: c[r] = D[8*half + r][idx]
#pragma unroll
    for (int r = 0; r < 8; ++r) {
        float v = c[r];
        if (RELU) v = fmaxf(v, 0.0f);
        C[(size_t)(tm + half * 8 + r) * Nc + tn + idx] = v;
    }
}

// ---------------- layer-3 tail (N=1 column, plain VALU) ----------------

// H3[i] = dot(G2[i, 0:32], W3[0:32])
__global__ void gcn_gemv32(const float* __restrict__ G2, const float* __restrict__ W3,
                           float* __restrict__ H3) {
    int i = blockIdx.x * blockDim.x + threadIdx.x;
    if (i < NN) {
        float s = 0.0f;
#pragma unroll
        for (int k = 0; k < 32; ++k) s += G2[i * 32 + k] * W3[k];
        H3[i] = s;
    }
}

// v[i] = b3 + sum_j A[i][j] * H3[j]
__global__ void gcn_aggv(const float* __restrict__ Amat, const float* __restrict__ H3,
                         const float* __restrict__ b3, float* __restrict__ v) {
    int i = blockIdx.x * blockDim.x + threadIdx.x;
    if (i < NN) {
        float s = b3[0];
        for (int j = 0; j < NN; ++j) s += Amat[(size_t)i * NN + j] * H3[j];
        v[i] = s;
    }
}

// ---------------- the bandwidth kernel: out = v @ W_lin + b_lin (512 MB stream) ----------------
// 4 outputs per thread (float4 coalesced), v staged in LDS once per block.
// W_lin is 512 MB read exactly once (> 192 MB L2): use non-temporal loads/stores
// so the one-pass stream doesn't churn the L2 (TH_NT cache policy, ISA 7.3).
__global__ void gcn_big_gemv(const float* __restrict__ v, const float* __restrict__ W_lin,
                             const float* __restrict__ b_lin, float* __restrict__ out) {
    __shared__ float vs[NN];
    const int t = threadIdx.x;  // 256 threads
    vs[t] = v[t];
    vs[t + 256] = v[t + 256];
    __syncthreads();

    const int m = (blockIdx.x * blockDim.x + t) * 4;   // 65536 threads * 4 = 262144 outputs
    v4f acc = *(const v4f*)(b_lin + m);
    for (int k = 0; k < NN; ++k) {
        v4f w = __builtin_nontemporal_load((const v4f*)(W_lin + (size_t)k * OUTSZ + m));
        float vk = vs[k];
        acc[0] += vk * w[0];
        acc[1] += vk * w[1];
        acc[2] += vk * w[2];
        acc[3] += vk * w[3];
    }
    __builtin_nontemporal_store(acc, (v4f*)(out + m));
}

// ---------------- launcher ----------------

extern "C" void kernel_launch(void* const* d_in, const int* in_sizes, int n_in,
                              void* d_out, int out_size, void* d_ws, size_t ws_size,
                              hipStream_t stream) {
    const float* x        = (const float*)d_in[0];
    const int*   eidx     = (const int*)d_in[1];       // [2, E] flat, int32
    const float* W1       = (const float*)d_in[2];
    const float* b1       = (const float*)d_in[3];
    const float* W2       = (const float*)d_in[4];
    const float* b2       = (const float*)d_in[5];
    const float* W3       = (const float*)d_in[6];
    const float* b3       = (const float*)d_in[7];
    const float* W_lin    = (const float*)d_in[8];
    const float* b_lin    = (const float*)d_in[9];
    float*       out      = (float*)d_out;

    const int E = in_sizes[1] / 2;
    const int* src = eidx;
    const int* dst = eidx + E;

    // workspace layout (floats)
    float* ws   = (float*)d_ws;
    float* dis  = ws;                    // 512        (first used as deg)
    float* Amat = dis + NN;              // 512*512
    float* H1   = Amat + NN * NN;        // 512*64
    float* G1   = H1 + NN * 64;          // 512*64
    float* H2   = G1 + NN * 64;          // 512*32
    float* G2   = H2 + NN * 32;          // 512*32
    float* H3   = G2 + NN * 32;          // 512
    float* vv   = H3 + NN;               // 512

    const int B = 256;

    // 1) degree (self-loop = 1.0 init), dis = rsqrt(deg)
    gcn_fill_f32<<<(NN + B - 1) / B, B, 0, stream>>>(dis, NN, 1.0f);
    gcn_deg_accum<<<(E + B - 1) / B, B, 0, stream>>>(dst, dis, E);
    gcn_rsqrt<<<(NN + B - 1) / B, B, 0, stream>>>(dis, NN);

    // 2) dense normalized adjacency
    gcn_fill_f32<<<(NN * NN + B - 1) / B, B, 0, stream>>>(Amat, NN * NN, 0.0f);
    gcn_scatterA<<<(E + B - 1) / B, B, 0, stream>>>(src, dst, dis, Amat, E);
    gcn_diagA<<<(NN + B - 1) / B, B, 0, stream>>>(dis, Amat);

    // 3) layer 1: G1 = relu(A @ (x @ W1) + b1)    [WMMA f32]
    wmma_gemm_f32<NN, 64, NN, false, false><<<(NN / 16) * (64 / 16), 32, 0, stream>>>(x, W1, nullptr, H1);
    wmma_gemm_f32<NN, 64, NN, true, true><<<(NN / 16) * (64 / 16), 32, 0, stream>>>(Amat, H1, b1, G1);

    // 4) layer 2: G2 = relu(A @ (G1 @ W2) + b2)   [WMMA f32]
    wmma_gemm_f32<NN, 32, 64, false, false><<<(NN / 16) * (32 / 16), 32, 0, stream>>>(G1, W2, nullptr, H2);
    wmma_gemm_f32<NN, 32, NN, true, true><<<(NN / 16) * (32 / 16), 32, 0, stream>>>(Amat, H2, b2, G2);

    // 5) layer 3 (output width 1): v = A @ (G2 @ W3) + b3
    gcn_gemv32<<<(NN + B - 1) / B, B, 0, stream>>>(G2, W3, H3);
    gcn_aggv<<<(NN + B - 1) / B, B, 0, stream>>>(Amat, H3, b3, vv);

    // 6) out = v @ W_lin + b_lin   (streams the 512 MB weight matrix — the runtime bound)
    gcn_big_gemv<<<OUTSZ / 4 / B, B, 0, stream>>>(vv, W_lin, b_lin, out);
}